// GraphSAGEEncoder_4209067950557
// MI455X (gfx1250) — compile-verified
//
#include <hip/hip_runtime.h>

#define N_NODES 50000
#define N_EDGES 800000
#define HID     128
#define KCAT    256          // 2*HID
#define N_LAYERS 3
#define TAU     1.0f
#define GEPS    1e-30f
#define ALPHA   0.5f
#define LN_EPS  1e-5f

typedef __attribute__((ext_vector_type(2))) float v2f;
typedef __attribute__((ext_vector_type(8))) float v8f;

// ---- order-preserving float <-> uint mapping (for atomicMax on fp32) ----
__device__ __forceinline__ unsigned mapf(float f) {
    unsigned u = __float_as_uint(f);
    return (u & 0x80000000u) ? ~u : (u | 0x80000000u);
}
__device__ __forceinline__ float unmapf(unsigned m) {
    unsigned u = (m & 0x80000000u) ? (m ^ 0x80000000u) : ~m;
    return __uint_as_float(u);
}
#define MAPPED_NEG_INF 0x007FFFFFu   // mapf(-inf)

// =====================  input projection: h = x @ Wi + bi  =====================
// grid: N_NODES/16 blocks of 256 threads (8 waves). wave w -> 16x16 tile at n0=w*16.
__global__ __launch_bounds__(256)
void k_input_proj(const float* __restrict__ x, const float* __restrict__ Wi,
                  const float* __restrict__ bi, float* __restrict__ h)
{
    const int lane = threadIdx.x & 31;
    const int wave = threadIdx.x >> 5;
    const int half = lane >> 4;          // 0 or 1
    const int mr   = lane & 15;          // A: row M ; B/D: col N
    const int m0   = blockIdx.x * 16;
    const int n0   = wave * 16;

    v8f acc = {};
    const float* __restrict__ arow = x + (size_t)(m0 + mr) * HID;
    for (int k = 0; k < HID; k += 4) {
        const int ka = k + 2 * half;                 // K for VGPR0; VGPR1 is ka+1
        v2f a; a.x = arow[ka]; a.y = arow[ka + 1];
        v2f b; b.x = Wi[(size_t)ka * HID + n0 + mr];
               b.y = Wi[(size_t)(ka + 1) * HID + n0 + mr];
        acc = __builtin_amdgcn_wmma_f32_16x16x4_f32(false, a, false, b,
                                                    (short)0, acc, false, false);
    }
    const float bias = bi[n0 + mr];
#pragma unroll
    for (int r = 0; r < 8; ++r) {
        const int m = m0 + r + 8 * half;             // D row for VGPR r
        h[(size_t)m * HID + n0 + mr] = acc[r] + bias;
    }
}

// =====================  degree =====================
__global__ void k_zero_deg(int* __restrict__ deg) {
    int t = blockIdx.x * blockDim.x + threadIdx.x;
    if (t < N_NODES) deg[t] = 0;
}
__global__ void k_deg(const int* __restrict__ edst, int* __restrict__ deg) {
    int e = blockIdx.x * blockDim.x + threadIdx.x;
    if (e < N_EDGES) atomicAdd(&deg[edst[e]], 1);
}

// =====================  per-layer scatter-logsumexp =====================
__global__ void k_init_scatter(unsigned* __restrict__ mxu, float* __restrict__ sum) {
    int t = blockIdx.x * blockDim.x + threadIdx.x;
    if (t < N_NODES * HID) { mxu[t] = MAPPED_NEG_INF; sum[t] = 0.0f; }
}

// one thread per (edge, 4-feature chunk): 32 threads / edge
__global__ __launch_bounds__(256)
void k_scatter_max(const float* __restrict__ h, const int* __restrict__ esrc,
                   const int* __restrict__ edst, unsigned* __restrict__ mxu)
{
    int t = blockIdx.x * blockDim.x + threadIdx.x;
    int e = t >> 5;
    if (e >= N_EDGES) return;
    const int f0 = (t & 31) * 4;
    const int s = esrc[e], d = edst[e];
    const float4 v = *(const float4*)(h + (size_t)s * HID + f0);
    unsigned* p = mxu + (size_t)d * HID + f0;
    atomicMax(p + 0, mapf(v.x));
    atomicMax(p + 1, mapf(v.y));
    atomicMax(p + 2, mapf(v.z));
    atomicMax(p + 3, mapf(v.w));
}

__global__ __launch_bounds__(256)
void k_scatter_expsum(const float* __restrict__ h, const int* __restrict__ esrc,
                      const int* __restrict__ edst, const unsigned* __restrict__ mxu,
                      float* __restrict__ sum)
{
    int t = blockIdx.x * blockDim.x + threadIdx.x;
    int e = t >> 5;
    if (e >= N_EDGES) return;
    const int f0 = (t & 31) * 4;
    const int s = esrc[e], d = edst[e];
    const float4 v = *(const float4*)(h + (size_t)s * HID + f0);
    const unsigned* mp = mxu + (size_t)d * HID + f0;
    float* sp = sum + (size_t)d * HID + f0;
    atomicAdd(sp + 0, expf(v.x - unmapf(mp[0])));
    atomicAdd(sp + 1, expf(v.y - unmapf(mp[1])));
    atomicAdd(sp + 2, expf(v.z - unmapf(mp[2])));
    atomicAdd(sp + 3, expf(v.w - unmapf(mp[3])));
}

// agg = deg>0 ? tau*(mx + log(max(sum,eps))) : 0   (written in place into `sum`)
__global__ void k_combine(const unsigned* __restrict__ mxu, float* __restrict__ sum,
                          const int* __restrict__ deg)
{
    int t = blockIdx.x * blockDim.x + threadIdx.x;
    if (t >= N_NODES * HID) return;
    float out = 0.0f;
    if (deg[t >> 7] > 0) {
        const float m = unmapf(mxu[t]);
        const float s = sum[t];
        out = TAU * (m + logf(fmaxf(s, GEPS)));
    }
    sum[t] = out;
}

// =====================  layer GEMM (K=256 via implicit concat) + LN + ReLU + residual ====
// grid: N_NODES/16 blocks of 256 threads; block computes rows [m0, m0+16), all 128 cols.
__global__ __launch_bounds__(256)
void k_layer_gemm_ln(const float* __restrict__ h, const float* __restrict__ agg,
                     const float* __restrict__ W,     // [256][128] row-major
                     const float* __restrict__ bl, const float* __restrict__ gamma,
                     const float* __restrict__ beta, float* __restrict__ hout)
{
    __shared__ float tile[16 * HID];
    __shared__ float s_mu[16], s_rs[16];

    const int lane = threadIdx.x & 31;
    const int wave = threadIdx.x >> 5;
    const int half = lane >> 4;
    const int mr   = lane & 15;
    const int m0   = blockIdx.x * 16;
    const int n0   = wave * 16;

    const float* __restrict__ arow0 = h   + (size_t)(m0 + mr) * HID;
    const float* __restrict__ arow1 = agg + (size_t)(m0 + mr) * HID;

    v8f acc = {};
    for (int k = 0; k < KCAT; k += 4) {
        const int ka = k + 2 * half;
        const float* ap = (ka < HID) ? (arow0 + ka) : (arow1 + (ka - HID));
        v2f a; a.x = ap[0]; a.y = ap[1];
        v2f b; b.x = W[(size_t)ka * HID + n0 + mr];
               b.y = W[(size_t)(ka + 1) * HID + n0 + mr];
        acc = __builtin_amdgcn_wmma_f32_16x16x4_f32(false, a, false, b,
                                                    (short)0, acc, false, false);
    }
    const float bias = bl[n0 + mr];
#pragma unroll
    for (int r = 0; r < 8; ++r)
        tile[(r + 8 * half) * HID + n0 + mr] = acc[r] + bias;
    __syncthreads();

    // per-row mean/var (16 rows, one thread each)
    if (threadIdx.x < 16) {
        const int row = threadIdx.x;
        float s = 0.0f;
        for (int j = 0; j < HID; ++j) s += tile[row * HID + j];
        const float mu = s * (1.0f / HID);
        float v = 0.0f;
        for (int j = 0; j < HID; ++j) {
            const float d = tile[row * HID + j] - mu;
            v += d * d;
        }
        s_mu[row] = mu;
        s_rs[row] = rsqrtf(v * (1.0f / HID) + LN_EPS);
    }
    __syncthreads();

    // normalize + affine + relu + residual; 256 threads: (row, 8 cols) each
    const int row = threadIdx.x >> 4;
    const int c0  = (threadIdx.x & 15) * 8;
    const float mu = s_mu[row], rs = s_rs[row];
    const int gm = m0 + row;
#pragma unroll
    for (int j = 0; j < 8; ++j) {
        const int c = c0 + j;
        float v = (tile[row * HID + c] - mu) * rs * gamma[c] + beta[c];
        v = fmaxf(v, 0.0f);
        hout[(size_t)gm * HID + c] = ALPHA * h[(size_t)gm * HID + c] + (1.0f - ALPHA) * v;
    }
}

// =====================  host launcher =====================
extern "C" void kernel_launch(void* const* d_in, const int* in_sizes, int n_in,
                              void* d_out, int out_size, void* d_ws, size_t ws_size,
                              hipStream_t stream)
{
    const float* x     = (const float*)d_in[0];
    const int*   esrc  = (const int*)  d_in[1];
    const int*   edst  = (const int*)  d_in[2];
    const float* Wi    = (const float*)d_in[3];
    const float* bi    = (const float*)d_in[4];
    const float* Wl    = (const float*)d_in[5];   // [3][256][128]
    const float* bl    = (const float*)d_in[6];   // [3][128]
    const float* gamma = (const float*)d_in[7];   // [3][128]
    const float* beta  = (const float*)d_in[8];   // [3][128]
    float* out = (float*)d_out;

    char* ws = (char*)d_ws;
    const size_t HB = (size_t)N_NODES * HID * sizeof(float);   // 25.6 MB
    float*    hA  = (float*)   (ws);
    float*    hB  = (float*)   (ws + 1 * HB);
    unsigned* mxu = (unsigned*)(ws + 2 * HB);
    float*    sum = (float*)   (ws + 3 * HB);
    int*      deg = (int*)     (ws + 4 * HB);

    const dim3 blk(256);
    const int nf = N_NODES * HID;                 // 6.4M
    const int st = N_EDGES * 32;                  // 25.6M scatter threads

    k_zero_deg<<<(N_NODES + 255) / 256, blk, 0, stream>>>(deg);
    k_deg<<<(N_EDGES + 255) / 256, blk, 0, stream>>>(edst, deg);
    k_input_proj<<<N_NODES / 16, blk, 0, stream>>>(x, Wi, bi, hA);

    float* hcur = hA;
    float* hnxt = hB;
    for (int i = 0; i < N_LAYERS; ++i) {
        k_init_scatter<<<(nf + 255) / 256, blk, 0, stream>>>(mxu, sum);
        k_scatter_max<<<(st + 255) / 256, blk, 0, stream>>>(hcur, esrc, edst, mxu);
        k_scatter_expsum<<<(st + 255) / 256, blk, 0, stream>>>(hcur, esrc, edst, mxu, sum);
        k_combine<<<(nf + 255) / 256, blk, 0, stream>>>(mxu, sum, deg);

        float* tgt = (i == N_LAYERS - 1) ? out : hnxt;
        k_layer_gemm_ln<<<N_NODES / 16, blk, 0, stream>>>(
            hcur, sum, Wl + (size_t)i * KCAT * HID,
            bl + i * HID, gamma + i * HID, beta + i * HID, tgt);
        hnxt = hcur;
        hcur = tgt;
    }
}